// attn_v3_57286273794068
// MI455X (gfx1250) — compile-verified
//
#include <hip/hip_runtime.h>
#include <math.h>

// ---------------------------------------------------------------------------
// Shapes
#define BATCH 4
#define CDIM  256
#define CP    64
#define HWDIM 4096          // 64*64
#define MRED_ROWS 576       // 9 groups of 64 per stream
// LDS tile strides (padded against bank conflicts, 16B aligned)
#define ALD 68
#define XLD 132

typedef float v2f __attribute__((ext_vector_type(2)));
typedef float v8f __attribute__((ext_vector_type(8)));

__device__ __forceinline__ v8f wmma4(v2f a, v2f b, v8f c) {
  // D = A(16x4 f32) * B(4x16 f32) + C(16x16 f32)   [V_WMMA_F32_16X16X4_F32]
  return __builtin_amdgcn_wmma_f32_16x16x4_f32(
      /*neg_a=*/false, a, /*neg_b=*/false, b,
      /*c_mod=*/(short)0, c, /*reuse_a=*/false, /*reuse_b=*/false);
}

// ---------------------------------------------------------------------------
__global__ void k_zero(float* __restrict__ p, int n) {
  int i = blockIdx.x * blockDim.x + threadIdx.x;
  if (i < n) p[i] = 0.0f;
}

// ---------------------------------------------------------------------------
// Mred[s] (576x256) : row group g of stream s is W_red[e] @ Wqkv[3s+t]
//   g in 0..2 : t=0 (Q-reduce), e = s + 3g
//   g in 3..5 : t=1 (K-reduce), e = 3s + (g-3)
//   g in 6..8 : t=2 (V-reduce), e = s + 3(g-6)
// One wave per 16x16 output tile; 27*4*16 = 1728 tiles.
__global__ __launch_bounds__(256) void k_build_mred(
    const float* __restrict__ Wqkv, const float* __restrict__ Wred,
    float* __restrict__ Mred) {
  int wave = (blockIdx.x * blockDim.x + threadIdx.x) >> 5;
  if (wave >= 27 * 4 * 16) return;
  int lane = threadIdx.x & 31;
  int half = lane >> 4, lrow = lane & 15;
  int mat = wave >> 6;           // 0..26
  int rest = wave & 63;
  int mtile = rest >> 4;         // 0..3
  int ntile = rest & 15;         // 0..15
  int s = mat / 9, g = mat % 9;
  int t, e;
  if (g < 3)      { t = 0; e = s + 3 * g;       }
  else if (g < 6) { t = 1; e = 3 * s + (g - 3); }
  else            { t = 2; e = s + 3 * (g - 6); }
  const float* Arow = Wred + (e * CP + mtile * 16 + lrow) * CDIM;
  const float* Bm   = Wqkv + (3 * s + t) * CDIM * CDIM;
  v8f acc = {};
  for (int k = 0; k < CDIM; k += 4) {
    int ka = k + 2 * half;
    v2f a;  a.x  = Arow[ka];
            a.y  = Arow[ka + 1];
    v2f bf; bf.x = Bm[ka * CDIM + ntile * 16 + lrow];
            bf.y = Bm[(ka + 1) * CDIM + ntile * 16 + lrow];
    acc = wmma4(a, bf, acc);
  }
  float* outp = Mred + s * MRED_ROWS * CDIM;
  int rowbase = g * 64 + mtile * 16;
#pragma unroll
  for (int r = 0; r < 8; ++r)
    outp[(rowbase + r + 8 * half) * CDIM + ntile * 16 + lrow] = acc[r];
}

// ---------------------------------------------------------------------------
// Pass 1: R = relu(Mred[s] @ x_s[b]); accumulate per-row sum and max.
// Block: 256 thr = 8 waves (4 M-rows x 2 N-cols); block tile 64M x 128N, Kc=64.
__global__ __launch_bounds__(256) void k_stats(
    const float* __restrict__ Mred,
    const float* __restrict__ t_in, const float* __restrict__ b_in,
    const float* __restrict__ m_in,
    float* __restrict__ sumbuf, unsigned int* __restrict__ maxbuf) {
  __shared__ float Alds[64 * ALD];
  __shared__ float Xlds[64 * XLD];
  int nb = blockIdx.x;                  // 0..31
  int mb = blockIdx.y;                  // 0..8
  int sb = blockIdx.z;                  // s*4+b
  int s = sb >> 2, bb = sb & 3;
  const float* X = (s == 0 ? t_in : (s == 1 ? b_in : m_in)) + bb * CDIM * HWDIM;
  const float* A = Mred + s * MRED_ROWS * CDIM + mb * 64 * CDIM;
  int tid = threadIdx.x;
  int wave = tid >> 5, lane = tid & 31;
  int half = lane >> 4, lrow = lane & 15;
  int wm = wave >> 1, wn = wave & 1;
  v8f acc0 = {}, acc1 = {}, acc2 = {}, acc3 = {};
  for (int kc = 0; kc < 4; ++kc) {
    {   // stage A 64x64
      int c4 = tid & 15, r0 = tid >> 4;
#pragma unroll
      for (int i = 0; i < 4; ++i) {
        int r = r0 + 16 * i;
        float4 v = *(const float4*)(A + r * CDIM + kc * 64 + c4 * 4);
        *(float4*)(Alds + r * ALD + c4 * 4) = v;
      }
    }
    {   // stage X 64x128
      int c4 = tid & 31, r0 = tid >> 5;
#pragma unroll
      for (int i = 0; i < 8; ++i) {
        int r = r0 + 8 * i;
        float4 v = *(const float4*)(X + (kc * 64 + r) * HWDIM + nb * 128 + c4 * 4);
        *(float4*)(Xlds + r * XLD + c4 * 4) = v;
      }
    }
    __syncthreads();
#pragma unroll
    for (int kk = 0; kk < 64; kk += 4) {
      int ka = kk + 2 * half;
      v2f a; a.x = Alds[(wm * 16 + lrow) * ALD + ka];
             a.y = Alds[(wm * 16 + lrow) * ALD + ka + 1];
      const float* x0 = Xlds + ka * XLD + wn * 64 + lrow;
      const float* x1 = Xlds + (ka + 1) * XLD + wn * 64 + lrow;
      v2f b0; b0.x = x0[0];  b0.y = x1[0];
      v2f b1; b1.x = x0[16]; b1.y = x1[16];
      v2f b2; b2.x = x0[32]; b2.y = x1[32];
      v2f b3; b3.x = x0[48]; b3.y = x1[48];
      acc0 = wmma4(a, b0, acc0);
      acc1 = wmma4(a, b1, acc1);
      acc2 = wmma4(a, b2, acc2);
      acc3 = wmma4(a, b3, acc3);
    }
    __syncthreads();
  }
  // relu + per-row reduce (row M = wm*16 + r + 8*half; cols across lrow/tiles)
  float sums[8], maxs[8];
#pragma unroll
  for (int r = 0; r < 8; ++r) {
    float v0 = fmaxf(acc0[r], 0.f), v1 = fmaxf(acc1[r], 0.f);
    float v2 = fmaxf(acc2[r], 0.f), v3 = fmaxf(acc3[r], 0.f);
    sums[r] = (v0 + v1) + (v2 + v3);
    maxs[r] = fmaxf(fmaxf(v0, v1), fmaxf(v2, v3));
  }
#pragma unroll
  for (int off = 8; off >= 1; off >>= 1) {
#pragma unroll
    for (int r = 0; r < 8; ++r) {
      sums[r] += __shfl_xor(sums[r], off, 32);
      maxs[r] = fmaxf(maxs[r], __shfl_xor(maxs[r], off, 32));
    }
  }
  if (lrow == 0) {
    int rowg = mb * 64 + wm * 16 + 8 * half;
    float* sp = sumbuf + (s * 4 + bb) * MRED_ROWS;
    unsigned int* mp = maxbuf + (s * 4 + bb) * MRED_ROWS;
#pragma unroll
    for (int r = 0; r < 8; ++r) {
      atomicAdd(sp + rowg + r, sums[r]);
      // relu outputs >= 0: IEEE bits order == uint order, init 0 is exact
      atomicMax(mp + rowg + r, __float_as_uint(maxs[r]));
    }
  }
}

// ---------------------------------------------------------------------------
// Per (head e, batch b): pools -> softmax over i -> vout -> o=sigmoid(W_rec@)
// -> accumulate alphaQ[e%3], alphaK[e/3].  36 blocks x 64 threads.
__global__ __launch_bounds__(64) void k_attn_small(
    const float* __restrict__ sumbuf, const unsigned int* __restrict__ maxbuf,
    const float* __restrict__ Wrec,
    float* __restrict__ alphaQ, float* __restrict__ alphaK) {
  __shared__ float fsh[64], gsh[64], hsh[64], vsh[64];
  int eb = blockIdx.x;
  int e = eb >> 2, bb = eb & 3;
  int i = threadIdx.x;
  int si = e / 3;   // IIDX[e]
  int sj = e % 3;   // JIDX[e]
  int kq = e / 3;   // q/v row-group index within stream sj
  int kk = e % 3;   // k  row-group index within stream si
  {
    int idx = (si * 4 + bb) * MRED_ROWS + 192 + 64 * kk + i;     // kr -> f
    fsh[i] = sumbuf[idx] * (1.f / 4096.f) + __uint_as_float(maxbuf[idx]);
    idx = (sj * 4 + bb) * MRED_ROWS + 64 * kq + i;               // qr -> g
    gsh[i] = sumbuf[idx] * (1.f / 4096.f) + __uint_as_float(maxbuf[idx]);
    idx = (sj * 4 + bb) * MRED_ROWS + 384 + 64 * kq + i;         // vr -> h
    hsh[i] = sumbuf[idx] * (1.f / 4096.f) + __uint_as_float(maxbuf[idx]);
  }
  __syncthreads();
  float gj = gsh[i];
  float mx = -3.4e38f;
  for (int ii = 0; ii < 64; ++ii) mx = fmaxf(mx, fsh[ii] * gj);
  float den = 0.f, num = 0.f;
  for (int ii = 0; ii < 64; ++ii) {
    float ex = __expf(fsh[ii] * gj - mx);
    den += ex;
    num += hsh[ii] * ex;
  }
  vsh[i] = num / den;
  __syncthreads();
  const float* Wr = Wrec + e * CDIM * CP;
#pragma unroll
  for (int q = 0; q < 4; ++q) {
    int c = i + 64 * q;
    float acc = 0.f;
    for (int e2 = 0; e2 < CP; ++e2) acc += vsh[e2] * Wr[c * CP + e2];
    float o = 1.f / (1.f + __expf(-acc));
    atomicAdd(alphaQ + (sj * 4 + bb) * CDIM + c, o);
    atomicAdd(alphaK + (si * 4 + bb) * CDIM + c, o);
  }
}

// ---------------------------------------------------------------------------
// Amat[b][s] = diag(alphaQ[s][b])*(W[3s]+W[3s+2]) + diag(alphaK[s][b])*W[3s+1]
__global__ __launch_bounds__(256) void k_build_amat(
    const float* __restrict__ Wqkv,
    const float* __restrict__ alphaQ, const float* __restrict__ alphaK,
    float* __restrict__ Amat) {
  int idx = blockIdx.x * blockDim.x + threadIdx.x;
  const int total = BATCH * 3 * CDIM * CDIM;
  if (idx >= total) return;
  int c2 = idx & 255;
  int c  = (idx >> 8) & 255;
  int s  = (idx >> 16) % 3;
  int bb = idx / (3 * CDIM * CDIM);
  float aq = alphaQ[(s * 4 + bb) * CDIM + c];
  float ak = alphaK[(s * 4 + bb) * CDIM + c];
  const float* W0 = Wqkv + (3 * s + 0) * CDIM * CDIM;
  const float* W1 = Wqkv + (3 * s + 1) * CDIM * CDIM;
  const float* W2 = Wqkv + (3 * s + 2) * CDIM * CDIM;
  Amat[idx] = aq * (W0[c * CDIM + c2] + W2[c * CDIM + c2]) + ak * W1[c * CDIM + c2];
}

// ---------------------------------------------------------------------------
// Pass 2: out[b] = sum_s Amat[b][s] @ x_s[b]   (256 x 4096, K = 3*256)
__global__ __launch_bounds__(256) void k_out(
    const float* __restrict__ Amat,
    const float* __restrict__ t_in, const float* __restrict__ b_in,
    const float* __restrict__ m_in, float* __restrict__ outp) {
  __shared__ float Alds[64 * ALD];
  __shared__ float Xlds[64 * XLD];
  int nb = blockIdx.x;     // 0..31
  int mb = blockIdx.y;     // 0..3
  int bb = blockIdx.z;     // 0..3
  int tid = threadIdx.x;
  int wave = tid >> 5, lane = tid & 31;
  int half = lane >> 4, lrow = lane & 15;
  int wm = wave >> 1, wn = wave & 1;
  v8f acc0 = {}, acc1 = {}, acc2 = {}, acc3 = {};
  for (int s = 0; s < 3; ++s) {
    const float* X = (s == 0 ? t_in : (s == 1 ? b_in : m_in)) + bb * CDIM * HWDIM;
    const float* A = Amat + (bb * 3 + s) * CDIM * CDIM + mb * 64 * CDIM;
    for (int kc = 0; kc < 4; ++kc) {
      {
        int c4 = tid & 15, r0 = tid >> 4;
#pragma unroll
        for (int i = 0; i < 4; ++i) {
          int r = r0 + 16 * i;
          float4 v = *(const float4*)(A + r * CDIM + kc * 64 + c4 * 4);
          *(float4*)(Alds + r * ALD + c4 * 4) = v;
        }
      }
      {
        int c4 = tid & 31, r0 = tid >> 5;
#pragma unroll
        for (int i = 0; i < 8; ++i) {
          int r = r0 + 8 * i;
          float4 v = *(const float4*)(X + (kc * 64 + r) * HWDIM + nb * 128 + c4 * 4);
          *(float4*)(Xlds + r * XLD + c4 * 4) = v;
        }
      }
      __syncthreads();
#pragma unroll
      for (int kk = 0; kk < 64; kk += 4) {
        int ka = kk + 2 * half;
        v2f a; a.x = Alds[(wm * 16 + lrow) * ALD + ka];
               a.y = Alds[(wm * 16 + lrow) * ALD + ka + 1];
        const float* x0 = Xlds + ka * XLD + wn * 64 + lrow;
        const float* x1 = Xlds + (ka + 1) * XLD + wn * 64 + lrow;
        v2f b0; b0.x = x0[0];  b0.y = x1[0];
        v2f b1; b1.x = x0[16]; b1.y = x1[16];
        v2f b2; b2.x = x0[32]; b2.y = x1[32];
        v2f b3; b3.x = x0[48]; b3.y = x1[48];
        acc0 = wmma4(a, b0, acc0);
        acc1 = wmma4(a, b1, acc1);
        acc2 = wmma4(a, b2, acc2);
        acc3 = wmma4(a, b3, acc3);
      }
      __syncthreads();
    }
  }
  float* O = outp + bb * CDIM * HWDIM + (mb * 64 + wm * 16) * HWDIM + nb * 128 + wn * 64;
#pragma unroll
  for (int r = 0; r < 8; ++r) {
    int row = r + 8 * half;
    O[row * HWDIM +  0 + lrow] = acc0[r];
    O[row * HWDIM + 16 + lrow] = acc1[r];
    O[row * HWDIM + 32 + lrow] = acc2[r];
    O[row * HWDIM + 48 + lrow] = acc3[r];
  }
}

// ---------------------------------------------------------------------------
extern "C" void kernel_launch(void* const* d_in, const int* in_sizes, int n_in,
                              void* d_out, int out_size, void* d_ws, size_t ws_size,
                              hipStream_t stream) {
  const float* t_in = (const float*)d_in[0];
  const float* b_in = (const float*)d_in[1];
  const float* m_in = (const float*)d_in[2];
  const float* Wqkv = (const float*)d_in[3];
  const float* Wred = (const float*)d_in[4];
  const float* Wrec = (const float*)d_in[5];
  float* outp = (float*)d_out;

  float* ws     = (float*)d_ws;
  float* Mred   = ws;                                  // 3*576*256 = 442368
  float* sumbuf = Mred + 3 * MRED_ROWS * CDIM;         // 6912
  float* maxbuf = sumbuf + 3 * BATCH * MRED_ROWS;      // 6912 (uint bits)
  float* alphaQ = maxbuf + 3 * BATCH * MRED_ROWS;      // 3072
  float* alphaK = alphaQ + 3 * BATCH * CDIM;           // 3072
  float* Amat   = alphaK + 3 * BATCH * CDIM;           // 786432

  // zero sum/max/alpha accumulators (19968 floats)
  {
    int n = 2 * 3 * BATCH * MRED_ROWS + 2 * 3 * BATCH * CDIM;
    k_zero<<<(n + 255) / 256, 256, 0, stream>>>(sumbuf, n);
  }
  // combined reduction matrices (27 small WMMA GEMMs)
  k_build_mred<<<216, 256, 0, stream>>>(Wqkv, Wred, Mred);
  // pass 1: fused GEMM + relu + mean/max pooling
  k_stats<<<dim3(32, 9, 12), 256, 0, stream>>>(
      Mred, t_in, b_in, m_in, sumbuf, (unsigned int*)maxbuf);
  // tiny attention core -> per-channel gates
  k_attn_small<<<36, 64, 0, stream>>>(
      sumbuf, (const unsigned int*)maxbuf, Wrec, alphaQ, alphaK);
  // fold gates into per-(batch,stream) 256x256 matrices
  {
    int n = BATCH * 3 * CDIM * CDIM;
    k_build_amat<<<(n + 255) / 256, 256, 0, stream>>>(Wqkv, alphaQ, alphaK, Amat);
  }
  // pass 2: output GEMM
  k_out<<<dim3(32, 4, 4), 256, 0, stream>>>(Amat, t_in, b_in, m_in, outp);
}